// DanModel_42219528519894
// MI455X (gfx1250) — compile-verified
//
#include <hip/hip_runtime.h>
#include <hip/hip_bf16.h>

typedef __attribute__((ext_vector_type(2))) float v2f;
typedef __attribute__((ext_vector_type(8))) float v8f;

#define B_SZ 4096
#define S_SZ 200
#define E_SZ 300
#define NP_SZ 304   // E padded up to a multiple of 16 (19 WMMA tiles, no predication)
#define T_SZ 2

// ---------------------------------------------------------------------------
// Kernel 0: pad W [E_SZ x E_SZ] -> Wp [NP_SZ x E_SZ], rows 300..303 zeroed.
// Lets every GEMM tile be full so the WMMA loop needs no masking.
// ---------------------------------------------------------------------------
__global__ void pad_weights_kernel(const float* __restrict__ W,
                                   float* __restrict__ Wp) {
  const int idx = blockIdx.x * blockDim.x + threadIdx.x;
  if (idx >= NP_SZ * E_SZ) return;
  const int n = idx / E_SZ;
  Wp[idx] = (n < E_SZ) ? W[idx] : 0.0f;
}

// ---------------------------------------------------------------------------
// Kernel 1: pooled[b, :] = mean_t emb[x[b,t], :]   (row stride NP_SZ)
// One block per batch row. Token ids staged in LDS; 4 independent row
// accumulators keep HBM loads in flight (this stage is the roofline limiter:
// ~1 GB of gathered rows @ 23.3 TB/s). Column j handled by thread j
// (coalesced 1200B row reads across 300 lanes, 10 waves/block for latency).
// ---------------------------------------------------------------------------
__global__ void gather_mean_kernel(const int* __restrict__ x,
                                   const float* __restrict__ emb,
                                   float* __restrict__ pooled) {
  __shared__ int toks[S_SZ];
  const int b = blockIdx.x;
  const int* xrow = x + (size_t)b * S_SZ;
  for (int t = threadIdx.x; t < S_SZ; t += blockDim.x) toks[t] = xrow[t];
  __syncthreads();

  const int j = threadIdx.x;
  if (j < E_SZ) {
    float s0 = 0.f, s1 = 0.f, s2 = 0.f, s3 = 0.f;
    int t = 0;
    for (; t + 4 <= S_SZ; t += 4) {
      const float* r0 = emb + (size_t)toks[t + 0] * E_SZ;
      const float* r1 = emb + (size_t)toks[t + 1] * E_SZ;
      const float* r2 = emb + (size_t)toks[t + 2] * E_SZ;
      const float* r3 = emb + (size_t)toks[t + 3] * E_SZ;
      s0 += r0[j];
      s1 += r1[j];
      s2 += r2[j];
      s3 += r3[j];
    }
    for (; t < S_SZ; ++t) s0 += emb[(size_t)toks[t] * E_SZ + j];
    pooled[(size_t)b * NP_SZ + j] = (s0 + s1 + s2 + s3) * (1.0f / (float)S_SZ);
  }
}

// ---------------------------------------------------------------------------
// Kernel 2/3: C = relu(A @ Wp^T)
//   A : [B_SZ x E_SZ] row-major with row stride NP_SZ
//   Wp: [NP_SZ x E_SZ] row-major (zero-padded rows), so B = Wp^T
//   C : [B_SZ x NP_SZ]
// One wave per 16x16 output tile using V_WMMA_F32_16X16X4_F32, K stepped by 4
// (75 steps). f32 WMMA keeps reference numerics; the GEMM (1.47 GFLOP total)
// is far from the bottleneck. With NP padding, the inner loop is exactly
// {2x global_load_b64, v_wmma} — no cndmask, no store guards.
//
// Layouts (wave32):
//   A (16x4, 2 VGPR): lane L -> (M = L%16, K = (L/16)*2 + i)
//   B (4x16, 2 VGPR): lane L -> (N = L%16, K = (L/16)*2 + i), B[k][n] = Wp[n][k]
//   C/D (8 VGPR):     lane L, vgpr i -> (M = i + 8*(L/16), N = L%16)
// ---------------------------------------------------------------------------
__global__ void gemm_wmma_f32_relu(const float* __restrict__ A,
                                   const float* __restrict__ Wp,
                                   float* __restrict__ C) {
  const int wave = threadIdx.x >> 5;
  const int lane = threadIdx.x & 31;
  const int waves_per_block = blockDim.x >> 5;

  const int tiles_n = NP_SZ >> 4;                 // 19
  const int tile = blockIdx.x * waves_per_block + wave;
  if (tile >= (B_SZ >> 4) * tiles_n) return;
  const int tm = tile / tiles_n;
  const int tn = tile % tiles_n;

  const int half = lane >> 4;                     // 0 or 1
  const int lm = lane & 15;

  const float* arow = A  + (size_t)(tm * 16 + lm) * NP_SZ + half * 2;
  const float* wrow = Wp + (size_t)(tn * 16 + lm) * E_SZ  + half * 2;

  v8f acc = {};
  for (int k0 = 0; k0 < E_SZ; k0 += 4) {          // 8B-aligned v2f loads
    v2f a = *(const v2f*)(arow + k0);
    v2f b = *(const v2f*)(wrow + k0);
    acc = __builtin_amdgcn_wmma_f32_16x16x4_f32(
        /*neg_a=*/false, a, /*neg_b=*/false, b,
        /*c_mod=*/(short)0, acc, /*reuse_a=*/false, /*reuse_b=*/false);
  }

  float* crow = C + (size_t)(tm * 16 + half * 8) * NP_SZ + tn * 16 + lm;
  #pragma unroll
  for (int i = 0; i < 8; ++i) {
    crow[(size_t)i * NP_SZ] = fmaxf(acc[i], 0.f);
  }
}

// ---------------------------------------------------------------------------
// Kernel 4: out[b, t] = dot(H[b,:E_SZ], w3[t,:]), T=2. One wave per batch
// row (H row stride NP_SZ), shuffle reduction (wave32).
// ---------------------------------------------------------------------------
__global__ void head_kernel(const float* __restrict__ H,
                            const float* __restrict__ w3,
                            float* __restrict__ out) {
  const int wave = threadIdx.x >> 5;
  const int lane = threadIdx.x & 31;
  const int b = blockIdx.x * (blockDim.x >> 5) + wave;
  if (b >= B_SZ) return;

  const float* h = H + (size_t)b * NP_SZ;
  float a0 = 0.f, a1 = 0.f;
  for (int k = lane; k < E_SZ; k += 32) {
    const float hv = h[k];
    a0 += hv * w3[k];
    a1 += hv * w3[E_SZ + k];
  }
  #pragma unroll
  for (int off = 16; off > 0; off >>= 1) {
    a0 += __shfl_xor(a0, off, 32);
    a1 += __shfl_xor(a1, off, 32);
  }
  if (lane == 0) {
    out[(size_t)b * T_SZ + 0] = a0;
    out[(size_t)b * T_SZ + 1] = a1;
  }
}

extern "C" void kernel_launch(void* const* d_in, const int* in_sizes, int n_in,
                              void* d_out, int out_size, void* d_ws, size_t ws_size,
                              hipStream_t stream) {
  const int*   x   = (const int*)d_in[0];
  const float* emb = (const float*)d_in[1];
  const float* w1  = (const float*)d_in[2];
  const float* w2  = (const float*)d_in[3];
  const float* w3  = (const float*)d_in[4];
  float* out = (float*)d_out;

  // Workspace layout (floats): buf0 | buf1 | w1p | w2p  (~10.7 MB total)
  float* buf0 = (float*)d_ws;
  float* buf1 = buf0 + (size_t)B_SZ * NP_SZ;
  float* w1p  = buf1 + (size_t)B_SZ * NP_SZ;
  float* w2p  = w1p  + (size_t)NP_SZ * E_SZ;

  // 0) zero-pad weights so all WMMA tiles are full
  const int pad_elems = NP_SZ * E_SZ;
  pad_weights_kernel<<<(pad_elems + 255) / 256, 256, 0, stream>>>(w1, w1p);
  pad_weights_kernel<<<(pad_elems + 255) / 256, 256, 0, stream>>>(w2, w2p);

  // 1) pooled = mean(emb[x]) -> buf0   (bandwidth-dominated stage)
  gather_mean_kernel<<<B_SZ, 320, 0, stream>>>(x, emb, buf0);

  // 2) h1 = relu(pooled @ w1^T) -> buf1
  const int tiles = (B_SZ / 16) * (NP_SZ / 16);   // 256 * 19 = 4864
  const int wpb = 8;                              // 8 waves (256 threads) / block
  const int blocks = (tiles + wpb - 1) / wpb;
  gemm_wmma_f32_relu<<<blocks, wpb * 32, 0, stream>>>(buf0, w1p, buf1);

  // 3) h2 = relu(h1 @ w2^T) -> buf0 (reuse)
  gemm_wmma_f32_relu<<<blocks, wpb * 32, 0, stream>>>(buf1, w2p, buf0);

  // 4) out = h2 @ w3^T   (4096 x 2)
  head_kernel<<<(B_SZ + 7) / 8, 256, 0, stream>>>(buf0, w3, out);
}